// MixtureOfExperts_89172111000183
// MI455X (gfx1250) — compile-verified
//
#include <hip/hip_runtime.h>
#include <math.h>

// MoE, top-2 of 8 experts. T=2048 tokens, D=512, H=1024.
// Sparse routed execution: fp32 WMMA (v_wmma_f32_16x16x4_f32) GEMMs,
// TDM gather (tensor_load_to_lds, gather mode) for routed-token X tiles.

typedef __attribute__((ext_vector_type(2))) float v2f;
typedef __attribute__((ext_vector_type(8))) float v8f;
typedef __attribute__((ext_vector_type(4))) unsigned int u32x4;
typedef __attribute__((ext_vector_type(8))) int i32x8;
typedef __attribute__((ext_vector_type(4))) int i32x4;

constexpr int T = 2048;
constexpr int D = 512;
constexpr int H = 1024;
constexpr int E = 8;
constexpr int TILES = T / 16;   // 128 token tiles per expert (max)
constexpr int XSTR = D + 4;     // TDM pad: +2 dwords per 256 dwords => row stride 516
constexpr int HSTR = H + 4;     // padded LDS stride (bank-conflict avoidance)

// ---------------------------------------------------------------- zero output
__global__ __launch_bounds__(256) void zero_out_kernel(float* __restrict__ out) {
    int i = blockIdx.x * 256 + threadIdx.x;       // grid covers T*D/4 float4s
    float4 z = make_float4(0.f, 0.f, 0.f, 0.f);
    reinterpret_cast<float4*>(out)[i] = z;
}

// ---------------------------------------------------------------- router
__global__ __launch_bounds__(256) void router_kernel(
    const float* __restrict__ x, const float* __restrict__ wr,
    const float* __restrict__ br, float* __restrict__ logits_out,
    float* __restrict__ gates) {
    __shared__ float wl[E * D];                   // 16 KB
    for (int i = threadIdx.x; i < E * D; i += 256) wl[i] = wr[i];
    __syncthreads();

    int t = blockIdx.x * 256 + threadIdx.x;
    float acc[E];
#pragma unroll
    for (int e = 0; e < E; ++e) acc[e] = br[e];

    const float4* xr = reinterpret_cast<const float4*>(x + (size_t)t * D);
    for (int d4 = 0; d4 < D / 4; ++d4) {
        float4 xv = xr[d4];
        int d = d4 * 4;
#pragma unroll
        for (int e = 0; e < E; ++e) {
            acc[e] += xv.x * wl[e * D + d]     + xv.y * wl[e * D + d + 1]
                    + xv.z * wl[e * D + d + 2] + xv.w * wl[e * D + d + 3];
        }
    }

    float mx = acc[0];
#pragma unroll
    for (int e = 1; e < E; ++e) mx = fmaxf(mx, acc[e]);
    float p[E];
#pragma unroll
    for (int e = 0; e < E; ++e) p[e] = expf(acc[e] - mx);

    int e0 = 0; float p0 = p[0];
#pragma unroll
    for (int e = 1; e < E; ++e) if (p[e] > p0) { p0 = p[e]; e0 = e; }
    int e1 = -1; float p1 = -1.f;
#pragma unroll
    for (int e = 0; e < E; ++e) if (e != e0 && p[e] > p1) { p1 = p[e]; e1 = e; }
    float inv = 1.f / (p0 + p1);                  // softmax denom cancels in top-2

#pragma unroll
    for (int e = 0; e < E; ++e) logits_out[(size_t)t * E + e] = acc[e];
#pragma unroll
    for (int e = 0; e < E; ++e)
        gates[(size_t)t * E + e] = (e == e0) ? p0 * inv : (e == e1) ? p1 * inv : 0.f;
}

// ---------------------------------------------------------------- compaction
// One wave32 per expert: deterministic ballot-prefix compaction (ascending
// token order => deterministic lists AND satisfies TDM gather index ordering).
__global__ __launch_bounds__(256) void compact_kernel(
    const float* __restrict__ gates, int* __restrict__ ecount,
    int* __restrict__ etok, float* __restrict__ egate) {
    int e  = threadIdx.x >> 5;
    int ln = threadIdx.x & 31;
    int cnt = 0;
    for (int t0 = 0; t0 < T; t0 += 32) {
        int t = t0 + ln;
        float g = gates[(size_t)t * E + e];
        bool pred = g > 0.f;
        unsigned mask = __builtin_amdgcn_ballot_w32(pred);
        int off = __popc(mask & ((1u << ln) - 1u));
        if (pred) {
            etok [e * T + cnt + off] = t;
            egate[e * T + cnt + off] = g;
        }
        cnt += __popc(mask);
    }
    if (ln == 0) ecount[e] = cnt;
}

// ---------------------------------------------------------------- expert MLP
__global__ __launch_bounds__(256) void moe_expert_kernel(
    const float* __restrict__ x,  const float* __restrict__ w1,
    const float* __restrict__ w2, const int* __restrict__ ecount,
    const int* __restrict__ etok, const float* __restrict__ egate,
    float* __restrict__ out) {
    int e  = blockIdx.x / TILES;
    int m0 = (blockIdx.x % TILES) * 16;
    int cnt = ecount[e];
    if (m0 >= cnt) return;                        // uniform exit: EXEC stays full

    __shared__ float Xl[16 * XSTR];               // 33 KB, filled by TDM gather
    __shared__ float Hl[16 * HSTR];               // 66 KB
    __shared__ int   tokS[16];
    __shared__ float gateS[16];

    int tid = threadIdx.x;
    if (tid < 16) {
        int gi = m0 + tid;
        int tok = 0; float g = 0.f;               // pad rows: gate 0 => adds 0.0
        if (gi < cnt) { tok = etok[e * T + gi]; g = egate[e * T + gi]; }
        tokS[tid] = tok; gateS[tid] = g;
    }
    __syncthreads();

    // ---- TDM gather: one tensor_load_to_lds pulls 16 indexed token rows
    // (512 f32 each) of x into Xl, padding +2 dwords every 256 dwords
    // (row stride 516 floats -> conflict-free A-fragment reads).
    if (tid < 32) {                               // wave 0 issues the DMA
        unsigned long long ga = (unsigned long long)(uintptr_t)x;
        unsigned int lds_off = (unsigned int)(uintptr_t)&Xl[0];  // addr[31:0] = LDS offset

        u32x4 g0;
        g0.x = 0x80000001u;                       // count=1 | gather_mode | 16b idx
        g0.y = lds_off;                           // lds_addr
        g0.z = (unsigned int)(ga & 0xffffffffu);  // global_addr[31:0]
        g0.w = (unsigned int)((ga >> 32) & 0x01ffffffu) | (2u << 30);  // addr[56:32], type=2

        i32x8 g1;
        g1[0] = (int)((2u << 16) | (1u << 20) | (7u << 22) | (1u << 25));
        //       data_size=4B | pad_enable | pad_interval=256dw | pad_amount=2dw
        g1[1] = (int)(512u << 16);                // tensor_dim0 = 512
        g1[2] = (int)(2048u << 16);               // tensor_dim1 = 2048 rows
        g1[3] = (int)(512u << 16);                // tile_dim0 = 512
        g1[4] = 16;                               // tile_dim1 = #valid indices
        g1[5] = 512;                              // tensor_dim0_stride = 512
        g1[6] = 0;
        g1[7] = 0;

        i32x4 g2, g3;                             // 16 x 16-bit row indices
#pragma unroll
        for (int q = 0; q < 4; ++q) g2[q] = tokS[2 * q]     | (tokS[2 * q + 1] << 16);
#pragma unroll
        for (int q = 0; q < 4; ++q) g3[q] = tokS[8 + 2 * q] | (tokS[8 + 2 * q + 1] << 16);

        i32x8 g4 = {};                            // unused trailing group (zero-filled)

        __builtin_amdgcn_tensor_load_to_lds(g0, g1, g2, g3, g4, 0);
        __builtin_amdgcn_s_wait_tensorcnt(0);
    }
    __syncthreads();                              // publish Xl to all 8 waves

    int wv = tid >> 5, ln = tid & 31;
    int p  = ln & 15;                             // row/col index within half-wave
    int kb = (ln >> 4) * 2;                       // lanes 0-15: K 0,1; lanes 16-31: K 2,3
    int mbase = (ln >> 4) * 8;                    // C/D rows: 0-7 (lo) / 8-15 (hi)

    const float* w1e = w1 + (size_t)e * H * D;
    const float* w2e = w2 + (size_t)e * D * H;

#pragma unroll                                    // pre-warm fc2 weight rows in L2
    for (int j = 0; j < 4; ++j)
        __builtin_prefetch(w2e + (size_t)(((wv * 4 + j) * 16) + p) * H, 0, 0);

    // ---- fc1: each wave owns 8 of the 64 H-tiles (K over D=512)
    for (int j = 0; j < 8; ++j) {
        int n0 = (wv * 8 + j) * 16;
        v8f acc = {};
        const float* ap = &Xl[p * XSTR + kb];
        const float* bp = w1e + (size_t)(n0 + p) * D + kb;
#pragma unroll 4
        for (int d0 = 0; d0 < 256; d0 += 4) {     // first half-row (pre-pad)
            v2f a = *reinterpret_cast<const v2f*>(ap + d0);
            v2f b = *reinterpret_cast<const v2f*>(bp + d0);
            acc = __builtin_amdgcn_wmma_f32_16x16x4_f32(
                false, a, false, b, (short)0, acc, false, false);
        }
#pragma unroll 4
        for (int d0 = 256; d0 < 512; d0 += 4) {   // second half-row (+2dw TDM pad)
            v2f a = *reinterpret_cast<const v2f*>(ap + d0 + 2);
            v2f b = *reinterpret_cast<const v2f*>(bp + d0);
            acc = __builtin_amdgcn_wmma_f32_16x16x4_f32(
                false, a, false, b, (short)0, acc, false, false);
        }
#pragma unroll
        for (int r = 0; r < 8; ++r) {             // exact GELU, stash to LDS
            float v = acc[r];
            v = 0.5f * v * (1.0f + erff(v * 0.70710678118654752f));
            Hl[(mbase + r) * HSTR + n0 + p] = v;
        }
    }
    __syncthreads();

    // ---- fc2: each wave owns 4 of the 32 D-tiles (K over H=1024)
    for (int j = 0; j < 4; ++j) {
        int n0 = (wv * 4 + j) * 16;
        v8f acc = {};
        const float* ap = &Hl[p * HSTR + kb];
        const float* bp = w2e + (size_t)(n0 + p) * H + kb;
#pragma unroll 4
        for (int h0 = 0; h0 < H; h0 += 4) {
            v2f a = *reinterpret_cast<const v2f*>(ap + h0);
            v2f b = *reinterpret_cast<const v2f*>(bp + h0);
            acc = __builtin_amdgcn_wmma_f32_16x16x4_f32(
                false, a, false, b, (short)0, acc, false, false);
        }
#pragma unroll
        for (int r = 0; r < 8; ++r) {             // gate-weighted scatter-add
            int m = mbase + r;                    // exactly 2 adds/element: commutative
            unsafeAtomicAdd(&out[(size_t)tokS[m] * D + n0 + p], acc[r] * gateS[m]);
        }
    }
}

// ---------------------------------------------------------------- launch
extern "C" void kernel_launch(void* const* d_in, const int* in_sizes, int n_in,
                              void* d_out, int out_size, void* d_ws, size_t ws_size,
                              hipStream_t stream) {
    const float* x  = (const float*)d_in[0];   // [T, D]
    const float* wr = (const float*)d_in[1];   // [E, D]
    const float* br = (const float*)d_in[2];   // [E]
    const float* w1 = (const float*)d_in[3];   // [E, H, D]
    const float* w2 = (const float*)d_in[4];   // [E, D, H]

    float* out    = (float*)d_out;             // [T, D] then [T, E] logits
    float* logits = out + (size_t)T * D;

    float* gates  = (float*)d_ws;              // [T, E]
    int*   ecount = (int*)(gates + (size_t)T * E);
    int*   etok   = ecount + 16;               // [E, T]
    float* egate  = (float*)(etok + (size_t)E * T);  // [E, T]

    zero_out_kernel<<<(T * D / 4) / 256, 256, 0, stream>>>(out);
    router_kernel<<<T / 256, 256, 0, stream>>>(x, wr, br, logits, gates);
    compact_kernel<<<1, 256, 0, stream>>>(gates, ecount, etok, egate);
    moe_expert_kernel<<<E * TILES, 256, 0, stream>>>(x, w1, w2, ecount, etok, egate, out);
}